// GATGFTFSharedEncoder_26422638805060
// MI455X (gfx1250) — compile-verified
//
#include <hip/hip_runtime.h>
#include <math.h>

#define H     64
#define NPN   16384      // part nodes
#define NTN   131072     // torque nodes
#define NFN   131072     // force nodes
#define NE    262144     // edges per type
#define NB    256        // graphs
#define NPART 64
#define NEG   0.2f
#define EPS   1e-5f

typedef __attribute__((ext_vector_type(2))) float v2f;
typedef __attribute__((ext_vector_type(8))) float v8f;

__device__ __forceinline__ void atomicMaxF(float* addr, float v) {
  if (v >= 0.f) atomicMax((int*)addr, __float_as_int(v));
  else          atomicMin((unsigned int*)addr, __float_as_uint(v));
}

// ---------------- WMMA GEMM: hs[N,64] = x[N,64] @ W[64,64]  (f32 WMMA 16x16x4)
// One wave per 16x16 output tile. A-frag (16x4 f32): lanes 0-15 M=0..15 K={0,1},
// lanes 16-31 K={2,3}. B-frag (4x16) mirrored with N on lanes. C/D: 8 VGPRs.
__global__ __launch_bounds__(32)
void k_gemm64(const float* __restrict__ x, const float* __restrict__ W,
              float* __restrict__ hs) {
  const int tile = blockIdx.x;
  const int row0 = (tile >> 2) << 4;   // node block
  const int col0 = (tile & 3) << 4;    // output-feature block
  const int lane = threadIdx.x;
  const int mn   = lane & 15;          // M for A, N for B
  const int koff = (lane >> 4) << 1;   // 0 or 2
  v8f c = {};
#pragma unroll
  for (int kb = 0; kb < 16; ++kb) {
    const int k = (kb << 2) + koff;
    v2f a, b;
    a.x = x[(row0 + mn) * 64 + k];
    a.y = x[(row0 + mn) * 64 + k + 1];
    b.x = W[k * 64 + col0 + mn];
    b.y = W[(k + 1) * 64 + col0 + mn];
    c = __builtin_amdgcn_wmma_f32_16x16x4_f32(false, a, false, b, (short)0, c,
                                              false, false);
  }
  const int half = lane >> 4;
#pragma unroll
  for (int r = 0; r < 8; ++r)
    hs[(row0 + r + 8 * half) * 64 + col0 + mn] = c[r];
}

// ---------------- helpers
__global__ void k_fill(float* __restrict__ p, float v, int n) {
  int i = blockIdx.x * blockDim.x + threadIdx.x;
  if (i < n) p[i] = v;
}

__global__ void k_init_bias(float* __restrict__ out, const float* __restrict__ b1,
                            const float* __restrict__ b2, int n) {
  int i = blockIdx.x * blockDim.x + threadIdx.x;
  if (i >= n) return;
  int h = i & 63;
  float v = b1[h];
  if (b2) v += b2[h];
  out[i] = v;
}

__global__ void k_relu(float* __restrict__ p, int n) {
  int i = blockIdx.x * blockDim.x + threadIdx.x;
  if (i < n) p[i] = fmaxf(p[i], 0.f);
}

// xp = concat(mass @ emb_parts_W, emb_state[idx])
__global__ void k_embed(const float* __restrict__ mass, const int* __restrict__ ps,
                        const float* __restrict__ embW, const float* __restrict__ embS,
                        float* __restrict__ xp) {
  int i = blockIdx.x * blockDim.x + threadIdx.x;
  if (i >= NPN * 64) return;
  int n = i >> 6, h = i & 63;
  if (h < 32) {
    xp[i] = mass[n] * embW[h];
  } else {
    int idx = ps[n * 2] + 2 * ps[n * 2 + 1];
    xp[i] = embS[idx * 32 + (h - 32)];
  }
}

// va[0:64] = Ws @ a_s ; va[64:128] = Wd @ a_d
__global__ __launch_bounds__(128)
void k_va(const float* __restrict__ Ws, const float* __restrict__ as,
          const float* __restrict__ Wd, const float* __restrict__ ad,
          float* __restrict__ va) {
  int t = threadIdx.x;
  const float* W = (t < 64) ? Ws : Wd;
  const float* a = (t < 64) ? as : ad;
  int k = t & 63;
  float s = 0.f;
  for (int i = 0; i < 64; ++i) s += W[k * 64 + i] * a[i];
  va[t] = s;
}

// alpha[n] = x[n,:] . va   (one wave per node)
__global__ __launch_bounds__(256)
void k_alpha(const float* __restrict__ x, const float* __restrict__ va,
             float* __restrict__ alpha, int N) {
  int wid  = (blockIdx.x * blockDim.x + threadIdx.x) >> 5;
  int lane = threadIdx.x & 31;
  if (wid >= N) return;
  float s = x[(size_t)wid * 64 + lane] * va[lane] +
            x[(size_t)wid * 64 + lane + 32] * va[lane + 32];
  for (int off = 16; off; off >>= 1) s += __shfl_down(s, off);
  if (lane == 0) alpha[wid] = s;
}

// pass1: e = lrelu(as[src]+ad[dst]); store; atomic max per dst
__global__ void k_e1(const int* __restrict__ es, const int* __restrict__ ed,
                     const float* __restrict__ as, const float* __restrict__ ad,
                     float* __restrict__ ev, float* __restrict__ dmax, int n) {
  int e = blockIdx.x * blockDim.x + threadIdx.x;
  if (e >= n) return;
  float v = as[es[e]] + ad[ed[e]];
  v = (v > 0.f) ? v : NEG * v;
  ev[e] = v;
  atomicMaxF(dmax + ed[e], v);
}

// pass2: ex = exp(e - max[dst]); store in-place; atomic sum per dst
__global__ void k_e2(const int* __restrict__ ed, float* __restrict__ ev,
                     const float* __restrict__ dmax, float* __restrict__ dsum, int n) {
  int e = blockIdx.x * blockDim.x + threadIdx.x;
  if (e >= n) return;
  int d = ed[e];
  float x = expf(ev[e] - dmax[d]);
  ev[e] = x;
  atomicAdd(dsum + d, x);
}

// pass3: out[dst,:] += (ex/sum[dst]) * hs[src,:]  (one wave per edge, 2 feats/lane)
__global__ __launch_bounds__(256)
void k_e3(const int* __restrict__ es, const int* __restrict__ ed,
          const float* __restrict__ ev, const float* __restrict__ dsum,
          const float* __restrict__ hs, float* __restrict__ out, int n) {
  int idx  = blockIdx.x * blockDim.x + threadIdx.x;
  int e    = idx >> 5;
  int lane = idx & 31;
  if (e >= n) return;
  int s = es[e], d = ed[e];
  float al = ev[e] / dsum[d];
  atomicAdd(out + (size_t)d * 64 + lane,       al * hs[(size_t)s * 64 + lane]);
  atomicAdd(out + (size_t)d * 64 + lane + 32,  al * hs[(size_t)s * 64 + lane + 32]);
}

// LayerNorm(64) + affine + [64,2] head (one wave per node)
__global__ __launch_bounds__(256)
void k_ln_outa(const float* __restrict__ ap, const float* __restrict__ g,
               const float* __restrict__ be, const float* __restrict__ Wa,
               const float* __restrict__ ba, float* __restrict__ ra, int N) {
  int wid  = (blockIdx.x * blockDim.x + threadIdx.x) >> 5;
  int lane = threadIdx.x & 31;
  if (wid >= N) return;
  float x0 = ap[(size_t)wid * 64 + lane];
  float x1 = ap[(size_t)wid * 64 + lane + 32];
  float s = x0 + x1;
  for (int off = 16; off; off >>= 1) s += __shfl_down(s, off);
  s = __shfl(s, 0);
  float mu = s * (1.f / 64.f);
  float d0 = x0 - mu, d1 = x1 - mu;
  float v = d0 * d0 + d1 * d1;
  for (int off = 16; off; off >>= 1) v += __shfl_down(v, off);
  v = __shfl(v, 0);
  float inv = rsqrtf(v * (1.f / 64.f) + EPS);
  float r0 = d0 * inv * g[lane] + be[lane];
  float r1 = d1 * inv * g[lane + 32] + be[lane + 32];
  float p0 = r0 * Wa[lane * 2 + 0] + r1 * Wa[(lane + 32) * 2 + 0];
  float p1 = r0 * Wa[lane * 2 + 1] + r1 * Wa[(lane + 32) * 2 + 1];
  for (int off = 16; off; off >>= 1) {
    p0 += __shfl_down(p0, off);
    p1 += __shfl_down(p1, off);
  }
  if (lane == 0) {
    ra[wid * 2 + 0] = p0 + ba[0];
    ra[wid * 2 + 1] = p1 + ba[1];
  }
}

__global__ void k_actmax(const float* __restrict__ ra, const int* __restrict__ batch,
                         float* __restrict__ rmax, int n2) {
  int i = blockIdx.x * blockDim.x + threadIdx.x;
  if (i >= n2) return;
  int n = i >> 1, c = i & 1;
  atomicMaxF(rmax + batch[n] * 2 + c, ra[i]);
}

__global__ void k_actexp(float* __restrict__ ra, const int* __restrict__ batch,
                         const float* __restrict__ rmax, float* __restrict__ rsum, int n2) {
  int i = blockIdx.x * blockDim.x + threadIdx.x;
  if (i >= n2) return;
  int n = i >> 1, c = i & 1;
  float x = expf(ra[i] - rmax[batch[n] * 2 + c]);
  ra[i] = x;
  atomicAdd(rsum + batch[n] * 2 + c, x);
}

__global__ void k_actout(const float* __restrict__ ra, const int* __restrict__ batch,
                         const int* __restrict__ pid, const float* __restrict__ rsum,
                         float* __restrict__ act, int n2) {
  int i = blockIdx.x * blockDim.x + threadIdx.x;
  if (i >= n2) return;
  int n = i >> 1, c = i & 1;
  int b = batch[n];
  act[b * 128 + c * 64 + pid[n]] = ra[i] / rsum[b * 2 + c];
}

// pooling: rep[b, type*192 + {0:max,64:min,128:mean} + h]
__global__ __launch_bounds__(64)
void k_pool(const float* __restrict__ xp, const float* __restrict__ xt,
            const float* __restrict__ xf, float* __restrict__ rep) {
  int b = blockIdx.x, ty = blockIdx.y, h = threadIdx.x;
  const float* x;
  int cnt;
  if (ty == 0)      { x = xp; cnt = NPART; }
  else if (ty == 1) { x = xt; cnt = 512; }
  else              { x = xf; cnt = 512; }
  const float* base = x + (size_t)b * cnt * 64 + h;
  float mx = -INFINITY, mn = INFINITY, sm = 0.f;
  for (int i = 0; i < cnt; ++i) {
    float v = base[(size_t)i * 64];
    mx = fmaxf(mx, v);
    mn = fminf(mn, v);
    sm += v;
  }
  float* r = rep + (size_t)b * 576 + ty * 192;
  r[h] = mx;
  r[64 + h] = mn;
  r[128 + h] = sm / (float)cnt;
}

__device__ __forceinline__ float gelu_exact(float x) {
  return 0.5f * x * (1.f + erff(x * 0.70710678118654752f));
}

// MLP: 576 -> 64 (gelu) -> 64 (gelu) -> 1 (tanh), one block per graph
__global__ __launch_bounds__(64)
void k_mlp(const float* __restrict__ rep, const float* __restrict__ W1,
           const float* __restrict__ b1, const float* __restrict__ W2,
           const float* __restrict__ b2, const float* __restrict__ W3,
           const float* __restrict__ b3, float* __restrict__ outV) {
  __shared__ float sin[576];
  __shared__ float h1[64];
  __shared__ float h2[64];
  int b = blockIdx.x, t = threadIdx.x;
  for (int i = t; i < 576; i += 64) sin[i] = rep[(size_t)b * 576 + i];
  __syncthreads();
  float s = b1[t];
  for (int k = 0; k < 576; ++k) s += sin[k] * W1[k * 64 + t];
  h1[t] = gelu_exact(s);
  __syncthreads();
  float s2 = b2[t];
  for (int k = 0; k < 64; ++k) s2 += h1[k] * W2[k * 64 + t];
  h2[t] = gelu_exact(s2);
  __syncthreads();
  if (t == 0) {
    float s3 = b3[0];
    for (int k = 0; k < 64; ++k) s3 += h2[k] * W3[k];
    outV[b] = tanhf(s3);
  }
}

extern "C" void kernel_launch(void* const* d_in, const int* in_sizes, int n_in,
                              void* d_out, int out_size, void* d_ws, size_t ws_size,
                              hipStream_t stream) {
  (void)in_sizes; (void)n_in; (void)out_size; (void)ws_size;
  const float* mass   = (const float*)d_in[0];
  const int*   pstate = (const int*)d_in[1];
  const float* torque = (const float*)d_in[2];
  const float* force  = (const float*)d_in[3];
  const int* es_[4] = {(const int*)d_in[4], (const int*)d_in[6],
                       (const int*)d_in[8], (const int*)d_in[10]};
  const int* ed_[4] = {(const int*)d_in[5], (const int*)d_in[7],
                       (const int*)d_in[9], (const int*)d_in[11]};
  const int* batch  = (const int*)d_in[12];
  const int* partid = (const int*)d_in[13];
  const float* embW = (const float*)d_in[14];
  const float* embS = (const float*)d_in[15];
  const float* Wsrc = (const float*)d_in[16];
  const float* Wdst = (const float*)d_in[17];
  const float* asrc = (const float*)d_in[18];
  const float* adst = (const float*)d_in[19];
  const float* bconv= (const float*)d_in[20];
  const float* lng  = (const float*)d_in[21];
  const float* lnb  = (const float*)d_in[22];
  const float* Wa   = (const float*)d_in[23];
  const float* ba   = (const float*)d_in[24];
  const float* W1   = (const float*)d_in[25];
  const float* b1   = (const float*)d_in[26];
  const float* W2   = (const float*)d_in[27];
  const float* b2   = (const float*)d_in[28];
  const float* W3   = (const float*)d_in[29];
  const float* b3   = (const float*)d_in[30];

  // ---- workspace layout (bump allocator)
  float* w = (float*)d_ws;
  auto alloc = [&](size_t n) { float* p = w; w += n; return p; };
  float* xpb[2] = {alloc((size_t)NPN * 64), alloc((size_t)NPN * 64)};
  float* xtb[2] = {alloc((size_t)NTN * 64), alloc((size_t)NTN * 64)};
  float* xfb[2] = {alloc((size_t)NFN * 64), alloc((size_t)NFN * 64)};
  float* apb    = alloc((size_t)NPN * 64);
  float* hs     = alloc((size_t)NTN * 64);   // max Ns
  float* al_s   = alloc(NTN);
  float* al_d   = alloc(NTN);
  float* dmax   = alloc(NTN);
  float* dsum   = alloc(NTN);
  float* ev     = alloc(NE);
  float* va     = alloc(128);
  float* ra     = alloc((size_t)NPN * 2);
  float* rmax   = alloc(512);
  float* rsum   = alloc(512);
  float* rep    = alloc((size_t)NB * 576);

  // ---- node embedding
  k_embed<<<(NPN * 64) / 256, 256, 0, stream>>>(mass, pstate, embW, embS, xpb[0]);

  const float* xp_c = xpb[0];
  const float* xt_c = torque;
  const float* xf_c = force;

  auto run_gat = [&](int l, int t, const float* xs, int Ns, const float* xd, int Nd,
                     float* outb) {
    const float* Ws = Wsrc + (size_t)(l * 4 + t) * 4096;
    const float* Wd = Wdst + (size_t)(l * 4 + t) * 4096;
    const float* as = asrc + (size_t)(l * 4 + t) * 64;
    const float* ad = adst + (size_t)(l * 4 + t) * 64;
    k_va<<<1, 128, 0, stream>>>(Ws, as, Wd, ad, va);
    k_gemm64<<<(Ns / 16) * 4, 32, 0, stream>>>(xs, Ws, hs);
    k_alpha<<<Ns / 8, 256, 0, stream>>>(xs, va, al_s, Ns);
    k_alpha<<<Nd / 8, 256, 0, stream>>>(xd, va + 64, al_d, Nd);
    k_fill<<<(Nd + 255) / 256, 256, 0, stream>>>(dmax, -INFINITY, Nd);
    k_fill<<<(Nd + 255) / 256, 256, 0, stream>>>(dsum, 0.f, Nd);
    k_e1<<<NE / 256, 256, 0, stream>>>(es_[t], ed_[t], al_s, al_d, ev, dmax, NE);
    k_e2<<<NE / 256, 256, 0, stream>>>(ed_[t], ev, dmax, dsum, NE);
    k_e3<<<NE / 8, 256, 0, stream>>>(es_[t], ed_[t], ev, dsum, hs, outb, NE);
  };

  // ---- 4 main hetero layers
  for (int l = 0; l < 4; ++l) {
    float* xp_n = (xp_c == xpb[0]) ? xpb[1] : xpb[0];
    float* xt_n = (xt_c == xtb[0]) ? xtb[1] : xtb[0];
    float* xf_n = (xf_c == xfb[0]) ? xfb[1] : xfb[0];
    k_init_bias<<<(NPN * 64) / 256, 256, 0, stream>>>(
        xp_n, bconv + (l * 4 + 1) * 64, bconv + (l * 4 + 3) * 64, NPN * 64);
    k_init_bias<<<(NTN * 64) / 256, 256, 0, stream>>>(
        xt_n, bconv + (l * 4 + 0) * 64, nullptr, NTN * 64);
    k_init_bias<<<(NFN * 64) / 256, 256, 0, stream>>>(
        xf_n, bconv + (l * 4 + 2) * 64, nullptr, NFN * 64);
    run_gat(l, 0, xp_c, NPN, xt_c, NTN, xt_n);   // p -> t
    run_gat(l, 1, xt_c, NTN, xp_c, NPN, xp_n);   // t -> p
    run_gat(l, 2, xp_c, NPN, xf_c, NFN, xf_n);   // p -> f
    run_gat(l, 3, xf_c, NFN, xp_c, NPN, xp_n);   // f -> p
    if (l < 3) {
      k_relu<<<(NPN * 64) / 256, 256, 0, stream>>>(xp_n, NPN * 64);
      k_relu<<<(NTN * 64) / 256, 256, 0, stream>>>(xt_n, NTN * 64);
      k_relu<<<(NFN * 64) / 256, 256, 0, stream>>>(xf_n, NFN * 64);
    }
    xp_c = xp_n; xt_c = xt_n; xf_c = xf_n;
  }

  // ---- actor layer: only out_p is consumed downstream (types 1 and 3)
  k_init_bias<<<(NPN * 64) / 256, 256, 0, stream>>>(
      apb, bconv + (4 * 4 + 1) * 64, bconv + (4 * 4 + 3) * 64, NPN * 64);
  run_gat(4, 1, xt_c, NTN, xp_c, NPN, apb);
  run_gat(4, 3, xf_c, NFN, xp_c, NPN, apb);

  // ---- LayerNorm + action head + per-graph softmax -> actions
  k_ln_outa<<<NPN / 8, 256, 0, stream>>>(apb, lng, lnb, Wa, ba, ra, NPN);
  k_fill<<<2, 256, 0, stream>>>(rmax, -INFINITY, 512);
  k_fill<<<2, 256, 0, stream>>>(rsum, 0.f, 512);
  k_actmax<<<(NPN * 2) / 256, 256, 0, stream>>>(ra, batch, rmax, NPN * 2);
  k_actexp<<<(NPN * 2) / 256, 256, 0, stream>>>(ra, batch, rmax, rsum, NPN * 2);
  float* out_act = (float*)d_out;
  k_fill<<<(NB * 128) / 256, 256, 0, stream>>>(out_act, 0.f, NB * 128);
  k_actout<<<(NPN * 2) / 256, 256, 0, stream>>>(ra, batch, partid, rsum, out_act,
                                                NPN * 2);

  // ---- critic: pool + MLP -> V
  k_pool<<<dim3(NB, 3), 64, 0, stream>>>(xp_c, xt_c, xf_c, rep);
  k_mlp<<<NB, 64, 0, stream>>>(rep, W1, b1, W2, b2, W3, b3, out_act + NB * 128);
}